// LSTM_704374636581
// MI455X (gfx1250) — compile-verified
//
#include <hip/hip_runtime.h>
#include <math.h>

// ---------------------------------------------------------------------------
// 2-layer LSTM (B=512, T=256, D=400, H=64) + FC head, fp32, CDNA5 WMMA + TDM.
//   Kernel 1: xg0 = x @ W_ih0^T + (b_ih0+b_hh0), stored time-major (T,B,4H).
//   Kernel 2: fused recurrence (both layers) + FC. h in LDS, c in registers,
//             xg0[t] tiles double-buffered into LDS by the Tensor Data Mover.
// ---------------------------------------------------------------------------

typedef __attribute__((ext_vector_type(2))) float v2f;   // f32 A/B frag (16x4 / 4x16)
typedef __attribute__((ext_vector_type(8))) float v8f;   // f32 C/D frag (16x16)
typedef __attribute__((ext_vector_type(4))) unsigned int u32x4;
typedef __attribute__((ext_vector_type(4))) int  i32x4;
typedef __attribute__((ext_vector_type(8))) int  i32x8;

#define Bsz  512
#define Tsz  256
#define Din  400
#define Hh   64
#define G4H  256          // 4*H

#ifndef __has_builtin
#define __has_builtin(x) 0
#endif
#if __has_builtin(__builtin_amdgcn_tensor_load_to_lds) && \
    __has_builtin(__builtin_amdgcn_s_wait_tensorcnt)
#define USE_TDM 1
#else
#define USE_TDM 0
#endif

// D = A(16x4) * B(4x16) + C, full fp32.
__device__ __forceinline__ v8f wmma4_f32(v2f a, v2f b, v8f c) {
    return __builtin_amdgcn_wmma_f32_16x16x4_f32(
        /*neg_a=*/false, a, /*neg_b=*/false, b,
        /*c_mod=*/(short)0, c, /*reuse_a=*/false, /*reuse_b=*/false);
}

__device__ __forceinline__ float sigmoidf_(float x) {
    return 1.0f / (1.0f + __expf(-x));
}

#if USE_TDM
// TDM: DMA a 2D tile (16 rows x 256 f32 cols, row stride 256) from global
// into LDS at byte offset lds_off. D# per CDNA5 ISA ch.8 (CDNA5 ISA).
__device__ __forceinline__ void tdm_load_tile(const float* gptr, unsigned lds_off) {
    const unsigned long long ga = (unsigned long long)(uintptr_t)gptr;
    u32x4 g0 = {};
    g0[0] = 1u;                                   // count=1, user load descriptor
    g0[1] = lds_off;                              // LDS byte address
    g0[2] = (unsigned)(ga & 0xFFFFFFFFu);         // global_addr[31:0]
    g0[3] = (unsigned)((ga >> 32) & 0x01FFFFFFu)  // global_addr[56:32]
          | 0x80000000u;                          // type = 2 ("image")
    i32x8 g1 = {};
    g1[0] = (int)(2u << 16);                      // workgroup_mask=0, data_size=4B
    g1[1] = (int)(256u << 16);                    // tensor_dim0 = 256 (gates)
    g1[2] = (int)(512u << 16);                    // tensor_dim1 = 512 (batch)
    g1[3] = (int)(256u << 16);                    // tile_dim0 = 256
    g1[4] = 16;                                   // tile_dim1 = 16, tile_dim2 = 0
    g1[5] = 256;                                  // tensor_dim0_stride[31:0]
    g1[6] = 0;                                    // stride hi / dim1_stride lo
    g1[7] = 0;
    i32x4 gz4 = {};
#if __clang_major__ >= 23
    i32x8 gz8 = {};
    __builtin_amdgcn_tensor_load_to_lds(g0, g1, gz4, gz4, gz8, 0);
#else
    __builtin_amdgcn_tensor_load_to_lds(g0, g1, gz4, gz4, 0);
#endif
}
#endif

// ---------------------------------------------------------------------------
// Kernel 1: input projection GEMM (unchanged WMMA pipeline).
//   xg0 (T,B,256): xg0[t][b][n] = sum_k x[b][t][k] * W_ih0[n][k] + bias[n]
// ---------------------------------------------------------------------------
#define LDA 401   // padded LDS row stride (odd -> conflict-free frag reads)

__global__ __launch_bounds__(256)
void lstm_xg0_gemm(const float* __restrict__ x,      // (B,T,400)
                   const float* __restrict__ Wih0,   // (256,400)
                   const float* __restrict__ bih0,   // (256,)
                   const float* __restrict__ bhh0,   // (256,)
                   float* __restrict__ xg0)          // (T,B,256)
{
    __shared__ float ldsA[16 * LDA];

    const int t     = blockIdx.x >> 5;
    const int btile = blockIdx.x & 31;
    const int b0    = btile * 16;
    const int tid   = threadIdx.x;

    for (int idx = tid; idx < 16 * Din; idx += 256) {
        int m = idx / Din, k = idx - m * Din;
        ldsA[m * LDA + k] = x[((size_t)(b0 + m) * Tsz + t) * Din + k];
    }
    __syncthreads();

    const int w    = tid >> 5;
    const int lane = tid & 31;
    const int half = lane >> 4;
    const int mrow = lane & 15;
    const int kh   = half * 2;

    const int n0 = w * 32, n1 = n0 + 16;
    v8f acc0 = {}; v8f acc1 = {};

    for (int ks = 0; ks < Din / 4; ++ks) {       // 100 k-steps of 4
        const int kb = ks * 4 + kh;
        v2f a;  a[0]  = ldsA[mrow * LDA + kb];
                a[1]  = ldsA[mrow * LDA + kb + 1];
        const float* wp0 = &Wih0[(size_t)(n0 + mrow) * Din + kb];
        const float* wp1 = &Wih0[(size_t)(n1 + mrow) * Din + kb];
        v2f bA; bA[0] = wp0[0]; bA[1] = wp0[1];
        v2f bB; bB[0] = wp1[0]; bB[1] = wp1[1];
        acc0 = wmma4_f32(a, bA, acc0);
        acc1 = wmma4_f32(a, bB, acc1);
    }

    const float bias0 = bih0[n0 + mrow] + bhh0[n0 + mrow];
    const float bias1 = bih0[n1 + mrow] + bhh0[n1 + mrow];
#pragma unroll
    for (int r = 0; r < 8; ++r) {
        const int m = r + half * 8;
        const size_t row = (size_t)t * Bsz + b0 + m;
        xg0[row * G4H + n0 + mrow] = acc0[r] + bias0;
        xg0[row * G4H + n1 + mrow] = acc1[r] + bias1;
    }
}

// ---------------------------------------------------------------------------
// Kernel 2: fused 2-layer recurrence + FC head.
// grid = B/16 blocks, 256 threads (8 waves). 16 batch rows per block for all
// timesteps; h0/h1 in LDS, c0/c1 in registers, xg0[t] tiles TDM-prefetched
// into a double-buffered LDS tile one step ahead of use.
// ---------------------------------------------------------------------------
#define LDH 65    // h LDS row stride (odd -> conflict-free A-frag reads)
#define LDG 257   // gate buffer row stride

__global__ __launch_bounds__(256)
void lstm_recurrence(const float* __restrict__ xg0,   // (T,B,256)
                     const float* __restrict__ Whh0,  // (256,64)
                     const float* __restrict__ Wih1,  // (256,64)
                     const float* __restrict__ Whh1,  // (256,64)
                     const float* __restrict__ bih1,  // (256,)
                     const float* __restrict__ bhh1,  // (256,)
                     const float* __restrict__ fcw,   // (400,64)
                     const float* __restrict__ fcb,   // (400,)
                     float* __restrict__ out)         // (B,400)
{
    __shared__ float h0[16 * LDH], h1[16 * LDH];
    __shared__ float gates[16 * LDG];
    __shared__ float bias1[G4H];
#if USE_TDM
    __shared__ float xbuf[2][16 * G4H];              // 2 x 16KB step tiles
#endif

    const int b0   = blockIdx.x * 16;
    const int tid  = threadIdx.x;
    const int w    = tid >> 5;
    const int lane = tid & 31;
    const int half = lane >> 4;
    const int mrow = lane & 15;
    const int kh   = half * 2;
    const int nA   = w * 32, nB = nA + 16;   // this wave's two N-tiles

    // Cell state in registers: thread owns (m,j) = (idx>>6, idx&63) for
    // idx = tid + 256*ii — identical mapping every timestep.
    float c0r[4] = {0.f, 0.f, 0.f, 0.f};
    float c1r[4] = {0.f, 0.f, 0.f, 0.f};

    for (int i = tid; i < 16 * LDH; i += 256) { h0[i] = 0.0f; h1[i] = 0.0f; }
    bias1[tid] = bih1[tid] + bhh1[tid];

#if USE_TDM
    if (w == 0) {
        tdm_load_tile(&xg0[(size_t)b0 * G4H],
                      (unsigned)(uintptr_t)&xbuf[0][0]);
        __builtin_amdgcn_s_wait_tensorcnt(0);
    }
#endif
    __syncthreads();

    for (int t = 0; t < Tsz; ++t) {
#if USE_TDM
        // Prefetch next timestep's gate pre-activations while computing t.
        if (w == 0 && t + 1 < Tsz) {
            tdm_load_tile(&xg0[((size_t)(t + 1) * Bsz + b0) * G4H],
                          (unsigned)(uintptr_t)&xbuf[(t + 1) & 1][0]);
        }
        const float* __restrict__ xr = &xbuf[t & 1][0];
        const int xstride = G4H;
#else
        const float* __restrict__ xr = &xg0[((size_t)t * Bsz + b0) * G4H];
        const int xstride = G4H;
#endif
        // ---- Layer 0 gate GEMM: gates = xg0[t] + h0 @ Whh0^T --------------
        {
            v8f acc0, acc1;
#pragma unroll
            for (int r = 0; r < 8; ++r) {
                const int m = r + half * 8;
                acc0[r] = xr[m * xstride + nA + mrow];
                acc1[r] = xr[m * xstride + nB + mrow];
            }
#pragma unroll
            for (int ks = 0; ks < Hh / 4; ++ks) {     // K = 64
                const int kb = ks * 4 + kh;
                v2f a;  a[0]  = h0[mrow * LDH + kb];
                        a[1]  = h0[mrow * LDH + kb + 1];
                v2f w0; w0[0] = Whh0[(nA + mrow) * Hh + kb];
                        w0[1] = Whh0[(nA + mrow) * Hh + kb + 1];
                v2f w1; w1[0] = Whh0[(nB + mrow) * Hh + kb];
                        w1[1] = Whh0[(nB + mrow) * Hh + kb + 1];
                acc0 = wmma4_f32(a, w0, acc0);
                acc1 = wmma4_f32(a, w1, acc1);
            }
#pragma unroll
            for (int r = 0; r < 8; ++r) {
                const int m = r + half * 8;
                gates[m * LDG + nA + mrow] = acc0[r];
                gates[m * LDG + nB + mrow] = acc1[r];
            }
        }
        __syncthreads();

        // ---- Layer 0 elementwise (gate order i,f,g,o): update c0(reg), h0 -
#pragma unroll
        for (int ii = 0; ii < 4; ++ii) {
            const int idx = tid + 256 * ii;
            const int m = idx >> 6, j = idx & 63;
            const float ig = sigmoidf_(gates[m * LDG + j]);
            const float fg = sigmoidf_(gates[m * LDG + 64 + j]);
            const float gg = tanhf    (gates[m * LDG + 128 + j]);
            const float og = sigmoidf_(gates[m * LDG + 192 + j]);
            const float c  = fg * c0r[ii] + ig * gg;
            c0r[ii] = c;
            h0[m * LDH + j] = og * tanhf(c);
        }
        __syncthreads();

        // ---- Layer 1 gate GEMM: gates = bias1 + h0 @ Wih1^T + h1 @ Whh1^T -
        {
            v8f acc0, acc1;
            const float bA = bias1[nA + mrow];
            const float bB = bias1[nB + mrow];
#pragma unroll
            for (int r = 0; r < 8; ++r) { acc0[r] = bA; acc1[r] = bB; }
#pragma unroll
            for (int ks = 0; ks < Hh / 4; ++ks) {     // hs0[t] contribution
                const int kb = ks * 4 + kh;
                v2f a;  a[0]  = h0[mrow * LDH + kb];
                        a[1]  = h0[mrow * LDH + kb + 1];
                v2f w0; w0[0] = Wih1[(nA + mrow) * Hh + kb];
                        w0[1] = Wih1[(nA + mrow) * Hh + kb + 1];
                v2f w1; w1[0] = Wih1[(nB + mrow) * Hh + kb];
                        w1[1] = Wih1[(nB + mrow) * Hh + kb + 1];
                acc0 = wmma4_f32(a, w0, acc0);
                acc1 = wmma4_f32(a, w1, acc1);
            }
#pragma unroll
            for (int ks = 0; ks < Hh / 4; ++ks) {     // h1(t-1) contribution
                const int kb = ks * 4 + kh;
                v2f a;  a[0]  = h1[mrow * LDH + kb];
                        a[1]  = h1[mrow * LDH + kb + 1];
                v2f w0; w0[0] = Whh1[(nA + mrow) * Hh + kb];
                        w0[1] = Whh1[(nA + mrow) * Hh + kb + 1];
                v2f w1; w1[0] = Whh1[(nB + mrow) * Hh + kb];
                        w1[1] = Whh1[(nB + mrow) * Hh + kb + 1];
                acc0 = wmma4_f32(a, w0, acc0);
                acc1 = wmma4_f32(a, w1, acc1);
            }
#pragma unroll
            for (int r = 0; r < 8; ++r) {
                const int m = r + half * 8;
                gates[m * LDG + nA + mrow] = acc0[r];
                gates[m * LDG + nB + mrow] = acc1[r];
            }
        }
        __syncthreads();

        // ---- Layer 1 elementwise: update c1(reg), h1 ----------------------
#pragma unroll
        for (int ii = 0; ii < 4; ++ii) {
            const int idx = tid + 256 * ii;
            const int m = idx >> 6, j = idx & 63;
            const float ig = sigmoidf_(gates[m * LDG + j]);
            const float fg = sigmoidf_(gates[m * LDG + 64 + j]);
            const float gg = tanhf    (gates[m * LDG + 128 + j]);
            const float og = sigmoidf_(gates[m * LDG + 192 + j]);
            const float c  = fg * c1r[ii] + ig * gg;
            c1r[ii] = c;
            h1[m * LDH + j] = og * tanhf(c);
        }
#if USE_TDM
        if (w == 0 && t + 1 < Tsz) {
            __builtin_amdgcn_s_wait_tensorcnt(0);   // next tile resident
        }
#endif
        __syncthreads();
    }

    // ---- FC head: out = h1(T-1) @ fcw^T + fcb  (400 cols = 25 tiles) ------
    for (int nt = w; nt < 25; nt += 8) {
        const int n = nt * 16 + mrow;
        v8f acc;
        const float bb = fcb[n];
#pragma unroll
        for (int r = 0; r < 8; ++r) acc[r] = bb;
#pragma unroll
        for (int ks = 0; ks < Hh / 4; ++ks) {
            const int kb = ks * 4 + kh;
            v2f a;  a[0]  = h1[mrow * LDH + kb];
                    a[1]  = h1[mrow * LDH + kb + 1];
            v2f wv; wv[0] = fcw[(size_t)n * Hh + kb];
                    wv[1] = fcw[(size_t)n * Hh + kb + 1];
            acc = wmma4_f32(a, wv, acc);
        }
#pragma unroll
        for (int r = 0; r < 8; ++r) {
            const int m = r + half * 8;
            out[(size_t)(b0 + m) * Din + n] = acc[r];
        }
    }
}

// ---------------------------------------------------------------------------
extern "C" void kernel_launch(void* const* d_in, const int* in_sizes, int n_in,
                              void* d_out, int out_size, void* d_ws, size_t ws_size,
                              hipStream_t stream) {
    const float* x    = (const float*)d_in[0];   // (512,256,4,100)
    const float* Wih0 = (const float*)d_in[1];   // (256,400)
    const float* Whh0 = (const float*)d_in[2];   // (256,64)
    const float* bih0 = (const float*)d_in[3];   // (256,)
    const float* bhh0 = (const float*)d_in[4];   // (256,)
    const float* Wih1 = (const float*)d_in[5];   // (256,64)
    const float* Whh1 = (const float*)d_in[6];   // (256,64)
    const float* bih1 = (const float*)d_in[7];   // (256,)
    const float* bhh1 = (const float*)d_in[8];   // (256,)
    const float* fcw  = (const float*)d_in[9];   // (400,64)
    const float* fcb  = (const float*)d_in[10];  // (400,)
    float* out = (float*)d_out;                  // (512,1,4,100) = (512,400)

    float* xg0 = (float*)d_ws;                   // (T,B,256) = 128 MiB scratch

    lstm_xg0_gemm<<<Tsz * (Bsz / 16), 256, 0, stream>>>(x, Wih0, bih0, bhh0, xg0);
    lstm_recurrence<<<Bsz / 16, 256, 0, stream>>>(xg0, Whh0, Wih1, Whh1,
                                                  bih1, bhh1, fcw, fcb, out);

    (void)in_sizes; (void)n_in; (void)out_size; (void)ws_size;
}